// TransformerEncoderBlock_3925600108666
// MI455X (gfx1250) — compile-verified
//
#include <hip/hip_runtime.h>
#include <math.h>

typedef __attribute__((ext_vector_type(2))) float v2f;
typedef __attribute__((ext_vector_type(8))) float v8f;

#define E_DIM 64
#define H_NUM 8
#define D_HEAD 8
#define B_SZ 16
#define S_LEN 1024
#define N_TOK (B_SZ * S_LEN)

// D = A(16x4 f32) * B(4x16 f32) + C(16x16 f32), full fp32 precision.
__device__ __forceinline__ v8f wmma4(v2f a, v2f b, v8f c) {
  return __builtin_amdgcn_wmma_f32_16x16x4_f32(
      /*neg_a=*/false, a, /*neg_b=*/false, b,
      /*c_mod=*/(short)0, c, /*reuse_a=*/false, /*reuse_b=*/false);
}

// ---------------------------------------------------------------------------
// Kernel 1: qkv = x @ Wqkv^T + bqkv      [16384,64] x [192,64]^T -> [16384,192]
// One 16x16 output tile per wave; 16 k-steps of WMMA f32 16x16x4.
// ---------------------------------------------------------------------------
__global__ __launch_bounds__(128) void k_qkv(const float* __restrict__ x,
                                             const float* __restrict__ Wqkv,
                                             const float* __restrict__ bqkv,
                                             float* __restrict__ qkv) {
  const int wave = threadIdx.x >> 5;
  const int lane = threadIdx.x & 31;
  const int n16 = lane & 15, hi = lane >> 4;
  const int tile = blockIdx.x * 4 + wave;        // 12288 tiles total
  const int mt = tile / 12, nt = tile % 12;
  const int m0 = mt * 16, n0 = nt * 16;
  const float* arow = x + (m0 + n16) * E_DIM;    // A: lane holds row m0+n16
  const float* brow = Wqkv + (n0 + n16) * E_DIM; // B(k,n)=Wqkv[n][k]
  v8f c = {};
#pragma unroll
  for (int k0 = 0; k0 < E_DIM; k0 += 4) {
    v2f a, b;
    a.x = arow[k0 + 2 * hi]; a.y = arow[k0 + 2 * hi + 1];
    b.x = brow[k0 + 2 * hi]; b.y = brow[k0 + 2 * hi + 1];
    c = wmma4(a, b, c);
  }
  const float bias = bqkv[n0 + n16];
#pragma unroll
  for (int r = 0; r < 8; ++r) {
    const int row = m0 + r + 8 * hi;             // C: vgpr r <-> rows r / r+8
    qkv[row * (3 * E_DIM) + n0 + n16] = c[r] + bias;
  }
}

// ---------------------------------------------------------------------------
// Kernel 2: flash attention. Block = 4 waves sharing one (batch, head);
// each wave owns one of 4 consecutive 16-query tiles. K/V tiles are staged
// cooperatively into double-buffered LDS (padded stride 9 -> bank-conflict
// free), with next-tile loads overlapping current-tile WMMAs. P is staged
// through wave-private LDS to convert C-layout -> A-layout for P@V.
// ---------------------------------------------------------------------------
__global__ __launch_bounds__(128) void k_attn(const float* __restrict__ qkv,
                                              float* __restrict__ attn_out) {
  __shared__ float ldsK[2][16 * 9];              // [buf][key*9 + dim]
  __shared__ float ldsV[2][16 * 9];
  __shared__ float ldsP[4][16 * 16];             // wave-private P staging
  const int tid = threadIdx.x;
  const int wave = tid >> 5;
  const int lane = tid & 31;
  const int n16 = lane & 15, hi = lane >> 4;

  const int bid = blockIdx.x;                    // 2048 blocks
  const int bh = bid >> 4;                       // 0..127 = (b,h)
  const int b = bh >> 3;
  const int h = bh & 7;
  const int qt = (bid & 15) * 4 + wave;          // query tile 0..63
  const float scale = 0.35355339059327373f;      // 1/sqrt(DH)

  const float* qkvB = qkv + (size_t)(b * S_LEN) * (3 * E_DIM);
  const int srow = tid >> 3, scol = tid & 7;     // staging: 16 rows x 8 dims

  // Q A-frags (scale folded in): K-dim = DH = 8 -> two k=4 chunks
  v2f qa0, qa1;
  {
    const float* qp = qkvB + (qt * 16 + n16) * (3 * E_DIM) + h * D_HEAD;
    qa0.x = qp[0 + 2 * hi] * scale; qa0.y = qp[1 + 2 * hi] * scale;
    qa1.x = qp[4 + 2 * hi] * scale; qa1.y = qp[5 + 2 * hi] * scale;
  }

  // Prologue: stage key tile 0 into buffer 0
  {
    const float* rowp = qkvB + srow * (3 * E_DIM) + h * D_HEAD + scol;
    ldsK[0][srow * 9 + scol] = rowp[E_DIM];          // K part
    ldsV[0][srow * 9 + scol] = rowp[2 * E_DIM];      // V part
  }

  float m[8], l[8];
  v8f o = {};
#pragma unroll
  for (int r = 0; r < 8; ++r) { m[r] = -3.0e38f; l[r] = 0.f; }
  float* myP = ldsP[wave];
  const int d = n16 & 7;                         // dup V cols 8..15 (unused)

  for (int kt = 0; kt < S_LEN / 16; ++kt) {
    __syncthreads();                             // buf[cur] staged; prev reads done
    const int cur = kt & 1, nxt = cur ^ 1;
    if (kt + 1 < S_LEN / 16) {                   // overlap next-tile loads
      const float* rowp =
          qkvB + ((kt + 1) * 16 + srow) * (3 * E_DIM) + h * D_HEAD + scol;
      ldsK[nxt][srow * 9 + scol] = rowp[E_DIM];
      ldsV[nxt][srow * 9 + scol] = rowp[2 * E_DIM];
    }

    // K B-frags from LDS: B(kdim, n=key) = K[key][kdim]
    v2f kb0, kb1;
    kb0.x = ldsK[cur][n16 * 9 + 0 + 2 * hi];
    kb0.y = ldsK[cur][n16 * 9 + 1 + 2 * hi];
    kb1.x = ldsK[cur][n16 * 9 + 4 + 2 * hi];
    kb1.y = ldsK[cur][n16 * 9 + 5 + 2 * hi];
    v8f s = {};
    s = wmma4(qa0, kb0, s);
    s = wmma4(qa1, kb1, s);

    // online softmax; row (r + 8*hi) lives in vgpr r across the lane half
#pragma unroll
    for (int r = 0; r < 8; ++r) {
      float mx = s[r];
      mx = fmaxf(mx, __shfl_xor(mx, 1));
      mx = fmaxf(mx, __shfl_xor(mx, 2));
      mx = fmaxf(mx, __shfl_xor(mx, 4));
      mx = fmaxf(mx, __shfl_xor(mx, 8));
      const float mn = fmaxf(m[r], mx);
      const float alpha = __expf(m[r] - mn);
      const float p = __expf(s[r] - mn);
      float ps = p;
      ps += __shfl_xor(ps, 1);
      ps += __shfl_xor(ps, 2);
      ps += __shfl_xor(ps, 4);
      ps += __shfl_xor(ps, 8);
      l[r] = l[r] * alpha + ps;
      m[r] = mn;
      o[r] *= alpha;
      myP[(r + 8 * hi) * 16 + n16] = p;          // C-layout -> LDS (m,n)
    }
    asm volatile("s_wait_dscnt 0" ::: "memory"); // wave-local LDS RAW (in-order DS)

    // O += P @ V : A = P (A-layout from LDS), B = V (from LDS), 4 k=4 chunks
#pragma unroll
    for (int kc = 0; kc < 4; ++kc) {
      v2f pa, vb;
      pa.x = myP[n16 * 16 + kc * 4 + 2 * hi];
      pa.y = myP[n16 * 16 + kc * 4 + 2 * hi + 1];
      vb.x = ldsV[cur][(kc * 4 + 2 * hi) * 9 + d];
      vb.y = ldsV[cur][(kc * 4 + 2 * hi + 1) * 9 + d];
      o = wmma4(pa, vb, o);
    }
    asm volatile("" ::: "memory");
  }

  if (n16 < D_HEAD) {
#pragma unroll
    for (int r = 0; r < 8; ++r) {
      const int row = b * S_LEN + qt * 16 + r + 8 * hi;
      attn_out[row * E_DIM + h * D_HEAD + n16] = o[r] / l[r];
    }
  }
}

// ---------------------------------------------------------------------------
// Kernel 3: mha_out = O @ Wout^T + bout; res1 = x + mha_out; mlp_in = LN1.
// Block computes full 16x64 row-block (4 waves x 16 cols) via LDS.
// ---------------------------------------------------------------------------
__global__ __launch_bounds__(128) void k_outproj_ln(
    const float* __restrict__ attn_out, const float* __restrict__ Wout,
    const float* __restrict__ bout, const float* __restrict__ x,
    const float* __restrict__ g1, const float* __restrict__ beta1,
    float* __restrict__ mlp_in) {
  __shared__ float res1[16][64];
  const int wave = threadIdx.x >> 5, lane = threadIdx.x & 31;
  const int n16 = lane & 15, hi = lane >> 4;
  const int m0 = blockIdx.x * 16;
  const int n0 = wave * 16;
  const float* arow = attn_out + (m0 + n16) * E_DIM;
  const float* brow = Wout + (n0 + n16) * E_DIM;
  v8f c = {};
#pragma unroll
  for (int k0 = 0; k0 < E_DIM; k0 += 4) {
    v2f a, b;
    a.x = arow[k0 + 2 * hi]; a.y = arow[k0 + 2 * hi + 1];
    b.x = brow[k0 + 2 * hi]; b.y = brow[k0 + 2 * hi + 1];
    c = wmma4(a, b, c);
  }
  const float bias = bout[n0 + n16];
#pragma unroll
  for (int r = 0; r < 8; ++r) {
    const int row = r + 8 * hi;
    res1[row][n0 + n16] = c[r] + bias + x[(m0 + row) * E_DIM + n0 + n16];
  }
  __syncthreads();
  // LayerNorm: 8 threads per row, 8 elements per thread
  const int row = threadIdx.x >> 3;
  const int c0 = (threadIdx.x & 7) * 8;
  float sum = 0.f, sq = 0.f;
#pragma unroll
  for (int i = 0; i < 8; ++i) { float v = res1[row][c0 + i]; sum += v; sq += v * v; }
#pragma unroll
  for (int msk = 1; msk < 8; msk <<= 1) {
    sum += __shfl_xor(sum, msk);
    sq += __shfl_xor(sq, msk);
  }
  const float mean = sum * (1.f / 64.f);
  const float var = sq * (1.f / 64.f) - mean * mean;
  const float rstd = rsqrtf(var + 1e-5f);
#pragma unroll
  for (int i = 0; i < 8; ++i) {
    const int col = c0 + i;
    mlp_in[(m0 + row) * E_DIM + col] =
        (res1[row][col] - mean) * rstd * g1[col] + beta1[col];
  }
}

// ---------------------------------------------------------------------------
// Kernel 4: h = relu(mlp_in @ Wm1^T + bm1)   [16384,64] x [128,64]^T
// ---------------------------------------------------------------------------
__global__ __launch_bounds__(128) void k_mlp1(const float* __restrict__ mlp_in,
                                              const float* __restrict__ Wm1,
                                              const float* __restrict__ bm1,
                                              float* __restrict__ hbuf) {
  const int wave = threadIdx.x >> 5, lane = threadIdx.x & 31;
  const int n16 = lane & 15, hi = lane >> 4;
  const int tile = blockIdx.x * 4 + wave;        // 8192 tiles
  const int mt = tile >> 3, nt = tile & 7;
  const int m0 = mt * 16, n0 = nt * 16;
  const float* arow = mlp_in + (m0 + n16) * E_DIM;
  const float* brow = Wm1 + (n0 + n16) * E_DIM;
  v8f c = {};
#pragma unroll
  for (int k0 = 0; k0 < E_DIM; k0 += 4) {
    v2f a, b;
    a.x = arow[k0 + 2 * hi]; a.y = arow[k0 + 2 * hi + 1];
    b.x = brow[k0 + 2 * hi]; b.y = brow[k0 + 2 * hi + 1];
    c = wmma4(a, b, c);
  }
  const float bias = bm1[n0 + n16];
#pragma unroll
  for (int r = 0; r < 8; ++r) {
    const int row = m0 + r + 8 * hi;
    hbuf[row * (2 * E_DIM) + n0 + n16] = fmaxf(c[r] + bias, 0.f);
  }
}

// ---------------------------------------------------------------------------
// Kernel 5: mlp_out = h @ Wm2^T + bm2; out = LN2(mlp_in + mlp_out)
// ---------------------------------------------------------------------------
__global__ __launch_bounds__(128) void k_mlp2_ln(
    const float* __restrict__ hbuf, const float* __restrict__ Wm2,
    const float* __restrict__ bm2, const float* __restrict__ mlp_in,
    const float* __restrict__ g2, const float* __restrict__ beta2,
    float* __restrict__ out) {
  __shared__ float res2[16][64];
  const int wave = threadIdx.x >> 5, lane = threadIdx.x & 31;
  const int n16 = lane & 15, hi = lane >> 4;
  const int m0 = blockIdx.x * 16;
  const int n0 = wave * 16;
  const float* arow = hbuf + (m0 + n16) * (2 * E_DIM);
  const float* brow = Wm2 + (n0 + n16) * (2 * E_DIM);
  v8f c = {};
#pragma unroll
  for (int k0 = 0; k0 < 2 * E_DIM; k0 += 4) {
    v2f a, b;
    a.x = arow[k0 + 2 * hi]; a.y = arow[k0 + 2 * hi + 1];
    b.x = brow[k0 + 2 * hi]; b.y = brow[k0 + 2 * hi + 1];
    c = wmma4(a, b, c);
  }
  const float bias = bm2[n0 + n16];
#pragma unroll
  for (int r = 0; r < 8; ++r) {
    const int row = r + 8 * hi;
    res2[row][n0 + n16] = c[r] + bias + mlp_in[(m0 + row) * E_DIM + n0 + n16];
  }
  __syncthreads();
  const int row = threadIdx.x >> 3;
  const int c0 = (threadIdx.x & 7) * 8;
  float sum = 0.f, sq = 0.f;
#pragma unroll
  for (int i = 0; i < 8; ++i) { float v = res2[row][c0 + i]; sum += v; sq += v * v; }
#pragma unroll
  for (int msk = 1; msk < 8; msk <<= 1) {
    sum += __shfl_xor(sum, msk);
    sq += __shfl_xor(sq, msk);
  }
  const float mean = sum * (1.f / 64.f);
  const float var = sq * (1.f / 64.f) - mean * mean;
  const float rstd = rsqrtf(var + 1e-5f);
#pragma unroll
  for (int i = 0; i < 8; ++i) {
    const int col = c0 + i;
    out[(m0 + row) * E_DIM + col] =
        (res2[row][col] - mean) * rstd * g2[col] + beta2[col];
  }
}

// ---------------------------------------------------------------------------
extern "C" void kernel_launch(void* const* d_in, const int* in_sizes, int n_in,
                              void* d_out, int out_size, void* d_ws, size_t ws_size,
                              hipStream_t stream) {
  const float* x    = (const float*)d_in[0];
  // d_in[1] coords, d_in[2] sizes, d_in[3] batch_size: layout is fixed
  const float* Wqkv = (const float*)d_in[4];
  const float* bqkv = (const float*)d_in[5];
  const float* Wout = (const float*)d_in[6];
  const float* bout = (const float*)d_in[7];
  const float* g1   = (const float*)d_in[8];
  const float* b1   = (const float*)d_in[9];
  const float* g2   = (const float*)d_in[10];
  const float* b2   = (const float*)d_in[11];
  const float* Wm1  = (const float*)d_in[12];
  const float* bm1  = (const float*)d_in[13];
  const float* Wm2  = (const float*)d_in[14];
  const float* bm2  = (const float*)d_in[15];
  float* out = (float*)d_out;

  float* ws      = (float*)d_ws;
  float* qkv     = ws;                                   // 16384*192
  float* attn    = qkv + (size_t)N_TOK * 3 * E_DIM;      // 16384*64
  float* mlpin   = attn + (size_t)N_TOK * E_DIM;         // 16384*64
  float* hbuf    = mlpin + (size_t)N_TOK * E_DIM;        // 16384*128

  k_qkv<<<(N_TOK / 16) * 12 / 4, 128, 0, stream>>>(x, Wqkv, bqkv, qkv);
  k_attn<<<(B_SZ * H_NUM * (S_LEN / 16)) / 4, 128, 0, stream>>>(qkv, attn);
  k_outproj_ln<<<N_TOK / 16, 128, 0, stream>>>(attn, Wout, bout, x, g1, b1, mlpin);
  k_mlp1<<<(N_TOK / 16) * 8 / 4, 128, 0, stream>>>(mlpin, Wm1, bm1, hbuf);
  k_mlp2_ln<<<N_TOK / 16, 128, 0, stream>>>(hbuf, Wm2, bm2, mlpin, g2, b2, out);
}